// AtlasMemoryLayer_33311766348071
// MI455X (gfx1250) — compile-verified
//
#include <hip/hip_runtime.h>
#include <math.h>
#include <stdint.h>

// ---------------- problem constants ----------------
constexpr int kB = 2, kT = 512, kC = 512, kH = 8, kD = 64, kE = 64;
constexpr int kCS = 64, kNC = kT / kCS;
constexpr int kHD = kH * kD;           // 512
constexpr float kGateBias = -2.0f;
constexpr float kMaxLR = 0.1f;
constexpr int kST = 68;                // LDS row stride (bank-conflict padding, 16B-aligned rows)
constexpr int kBUF = 64 * kST;         // one 64x64 LDS matrix buffer (floats)

__constant__ float PE_A[5] = {8.156554524902461f, 4.042929935166739f, 3.8916678022926607f,
                              3.285753657755655f, 2.3465413258596377f};
__constant__ float PE_B[5] = {-22.48329292557795f, -2.808917465908714f, -2.772484153217685f,
                              -2.3681294933425376f, -1.7097828382687081f};
__constant__ float PE_C[5] = {15.878769915207462f, 0.5000178451051316f, 0.5060648178503393f,
                              0.46449024233003106f, 0.42323551169305323f};

typedef float v2f __attribute__((ext_vector_type(2)));
typedef float v4f __attribute__((ext_vector_type(4)));
typedef float v8f __attribute__((ext_vector_type(8)));
typedef int v4i __attribute__((vector_size(4 * sizeof(int))));  // matches builtin proto

// ------------------------------------------------------------------
// CDNA5 async global->LDS copy (ASYNCcnt path). The clang builtin
// expects (v4i addrspace(1)*, v4i addrspace(3)*, imm offset, imm cpol).
// Fall back to inline asm when the builtin is unavailable.
// ------------------------------------------------------------------
#if defined(__gfx1250__) && __has_builtin(__builtin_amdgcn_global_load_async_to_lds_b128)
#define ATLAS_ASYNC_BUILTIN 1
#else
#define ATLAS_ASYNC_BUILTIN 0
#endif

__device__ __forceinline__ void copy16_async(const float* gsrc, float* ldst) {
#if ATLAS_ASYNC_BUILTIN
  __builtin_amdgcn_global_load_async_to_lds_b128(
      (__attribute__((address_space(1))) v4i*)gsrc,
      (__attribute__((address_space(3))) v4i*)ldst, 0, 0);
#else
  asm volatile("global_load_async_to_lds_b128 %0, %1, off"
               :
               : "v"((uint32_t)(uintptr_t)ldst), "v"((unsigned long long)(uintptr_t)gsrc)
               : "memory");
#endif
}

__device__ __forceinline__ void async_join() {
#if defined(__gfx1250__) && __has_builtin(__builtin_amdgcn_s_wait_asynccnt)
  __builtin_amdgcn_s_wait_asynccnt(0);
#else
  asm volatile("s_wait_asynccnt 0x0" ::: "memory");
#endif
}

// Async-load a 64x64 f32 tile (row stride rstride floats) into an LDS
// matrix buffer with row stride kST. 4 x b128 per thread, 256 threads.
__device__ __forceinline__ void tile_load_async64(const float* g0, size_t rstride,
                                                  float* ldsBuf, int tid) {
#pragma unroll
  for (int i = 0; i < 4; ++i) {
    int fidx = (tid + 256 * i) * 4;
    int r = fidx >> 6, c = fidx & 63;
    copy16_async(g0 + (size_t)r * rstride + c, ldsBuf + r * kST + c);
  }
}

__device__ __forceinline__ float gelu_f(float x) {
  float x3 = x * x * x;
  return 0.5f * x * (1.0f + tanhf(0.7978845608028654f * (x + 0.044715f * x3)));
}
__device__ __forceinline__ float gelu_deriv_f(float x) {
  float cdf = 0.5f * (1.0f + erff(x * 0.7071067811865476f));
  float pdf = expf(-0.5f * x * x) * 0.3989422804014327f;
  return cdf + x * pdf;
}

// ------------------------------------------------------------------
// 64x64x64 fp32 matmul in one 256-thread workgroup with
// V_WMMA_F32_16X16X4_F32. Operands in LDS (row stride kST).
// D = A*B (+ beta*C); transB reads B[n][k] (i.e. D = A*B^T).
// Fragments are preloaded into registers so the 16-WMMA chain issues
// back-to-back with a single dscnt wait.
// ------------------------------------------------------------------
__device__ __forceinline__ void mm64(const float* As, const float* Bs, float* Ds,
                                     const float* Cs, float beta, bool transB) {
  const int tid  = threadIdx.x;
  const int wave = tid >> 5;
  const int lane = tid & 31;
  const int tm   = wave & 3;            // tile row
  const int tn0  = (wave >> 2) * 2;     // tile col base (0 or 2)
  const int l16  = lane & 15;
  const int hi   = lane >> 4;
  const int arow = tm * 16 + l16;

  v2f a[16];
#pragma unroll
  for (int kk = 0; kk < 16; ++kk)
    a[kk] = *(const v2f*)(As + arow * kST + kk * 4 + hi * 2);

#pragma unroll
  for (int s = 0; s < 2; ++s) {
    const int tn = tn0 + s;
    const int bcol = tn * 16 + l16;
    v2f b[16];
#pragma unroll
    for (int kk = 0; kk < 16; ++kk) {
      const int k0 = kk * 4 + hi * 2;
      if (transB) {
        b[kk] = *(const v2f*)(Bs + bcol * kST + k0);
      } else {
        b[kk].x = Bs[k0 * kST + bcol];
        b[kk].y = Bs[(k0 + 1) * kST + bcol];
      }
    }
    v8f acc;
    if (Cs) {
#pragma unroll
      for (int v = 0; v < 8; ++v)
        acc[v] = beta * Cs[(tm * 16 + v + 8 * hi) * kST + (tn * 16 + l16)];
    } else {
#pragma unroll
      for (int v = 0; v < 8; ++v) acc[v] = 0.0f;
    }
#pragma unroll
    for (int kk = 0; kk < 16; ++kk)
      acc = __builtin_amdgcn_wmma_f32_16x16x4_f32(false, a[kk], false, b[kk], (short)0,
                                                  acc, false, false);
#pragma unroll
    for (int v = 0; v < 8; ++v)
      Ds[(tm * 16 + v + 8 * hi) * kST + (tn * 16 + l16)] = acc[v];
  }
}

// ------------------------------------------------------------------
// GEMM: O[M,N] = X[M,K] @ W[N,K]^T  (fp32; M,N,K multiples of 64).
// Double-buffered: async global->LDS copies of the next K-tile overlap
// the WMMA chain on the current tile. grid=(M/64,N/64), block=256,
// dynamic LDS = 4*kBUF floats. (Ping-pong pointers computed from `cur`;
// no LDS pointer aggregates -> no addrspacecast static initializers.)
// ------------------------------------------------------------------
__global__ __launch_bounds__(256) void gemm_xwt_kernel(const float* __restrict__ X,
                                                       const float* __restrict__ W,
                                                       float* __restrict__ O,
                                                       int M, int N, int K) {
  extern __shared__ float lds[];
  const int bm = blockIdx.x, bn = blockIdx.y;
  const int tid = threadIdx.x;
  const int wave = tid >> 5, lane = tid & 31;
  const int tm = wave & 3, tn0 = (wave >> 2) * 2;
  const int l16 = lane & 15, hi = lane >> 4;
  const int arow = tm * 16 + l16;

  const float* Xrow = X + (size_t)(bm * 64) * K;
  const float* Wrow = W + (size_t)(bn * 64) * K;

  v8f acc0, acc1;
#pragma unroll
  for (int v = 0; v < 8; ++v) { acc0[v] = 0.0f; acc1[v] = 0.0f; }

  // prologue: fetch first K-tile into buffers 0 (A) and 2 (B)
  tile_load_async64(Xrow, K, lds, tid);
  tile_load_async64(Wrow, K, lds + 2 * kBUF, tid);
  async_join();
  __syncthreads();

  int cur = 0;
  for (int kb = 0; kb < K; kb += 64) {
    float* As = lds + cur * kBUF;
    float* Bs = lds + (2 + cur) * kBUF;
    // prefetch next tile into the other buffers (overlaps the WMMAs)
    if (kb + 64 < K) {
      float* An = lds + (cur ^ 1) * kBUF;
      float* Bn = lds + (2 + (cur ^ 1)) * kBUF;
      tile_load_async64(Xrow + kb + 64, K, An, tid);
      tile_load_async64(Wrow + kb + 64, K, Bn, tid);
    }

    v2f a[16];
#pragma unroll
    for (int kk = 0; kk < 16; ++kk)
      a[kk] = *(const v2f*)(As + arow * kST + kk * 4 + hi * 2);
    v2f b0[16], b1[16];
#pragma unroll
    for (int kk = 0; kk < 16; ++kk) {
      const int k0 = kk * 4 + hi * 2;
      b0[kk] = *(const v2f*)(Bs + (tn0 * 16 + l16) * kST + k0);
      b1[kk] = *(const v2f*)(Bs + ((tn0 + 1) * 16 + l16) * kST + k0);
    }
#pragma unroll
    for (int kk = 0; kk < 16; ++kk) {
      acc0 = __builtin_amdgcn_wmma_f32_16x16x4_f32(false, a[kk], false, b0[kk], (short)0,
                                                   acc0, false, false);
      acc1 = __builtin_amdgcn_wmma_f32_16x16x4_f32(false, a[kk], false, b1[kk], (short)0,
                                                   acc1, false, false);
    }
    async_join();      // our prefetch writes have landed
    __syncthreads();   // everyone done reading cur + done writing nxt
    cur ^= 1;
  }
#pragma unroll
  for (int v = 0; v < 8; ++v) {
    int row = bm * 64 + tm * 16 + v + 8 * hi;
    O[(size_t)row * N + bn * 64 + tn0 * 16 + l16] = acc0[v];
    O[(size_t)row * N + bn * 64 + (tn0 + 1) * 16 + l16] = acc1[v];
  }
}

// ------------------------------------------------------------------
// Causal depthwise conv(K=4) + optional poly feature + RMS norm.
// grid = B*T*H, block = 64.
// ------------------------------------------------------------------
__global__ void convhead_kernel(const float* __restrict__ raw, const float* __restrict__ cw,
                                const float* __restrict__ cb, const float* __restrict__ poly,
                                float* __restrict__ out, int do_poly) {
  const int bid = blockIdx.x;
  const int hh = bid % kH;
  const int t = (bid / kH) % kT;
  const int b = bid / (kH * kT);
  const int d = threadIdx.x;
  const int ch = hh * 64 + d;

  float acc = cb[ch];
#pragma unroll
  for (int j = 0; j < 4; ++j) {
    int tt = t - 3 + j;
    if (tt >= 0) acc += cw[ch * 4 + j] * raw[((size_t)(b * kT + tt)) * kHD + ch];
  }
  float r = acc;
  __shared__ float red[64];
  if (do_poly) {
    r = poly[0] * acc + poly[1] * acc * acc;
    red[d] = r * r;
    __syncthreads();
#pragma unroll
    for (int s = 32; s > 0; s >>= 1) {
      if (d < s) red[d] += red[d + s];
      __syncthreads();
    }
    float ms = red[0] * (1.0f / 64.0f);
    r = r * rsqrtf(ms + 1e-6f);
  }
  out[((size_t)(b * kT + t) * kH + hh) * 64 + d] = r;
}

// ------------------------------------------------------------------
// All four sigmoid gates. grid = B*T, block = 256 (wave w -> head w).
// ------------------------------------------------------------------
__global__ void gates_kernel(const float* __restrict__ x,
                             const float* aw, const float* ab, const float* ew,
                             const float* eb, const float* tw, const float* tb,
                             const float* gw, const float* gb,
                             float* alpha, float* eta, float* theta, float* gamma) {
  const int t = blockIdx.x;
  __shared__ __attribute__((aligned(16))) float xs[512];
  const int tid = threadIdx.x;
  *(v4f*)&xs[tid * 2] = *(const v4f*)&x[(size_t)t * kC + tid * 2];
  __syncthreads();
  const int wave = tid >> 5, lane = tid & 31;
  const float* Ws[4] = {aw, ew, tw, gw};
  const float* Bs4[4] = {ab, eb, tb, gb};
  float* Os[4] = {alpha, eta, theta, gamma};
#pragma unroll
  for (int g = 0; g < 4; ++g) {
    const float* wrow = Ws[g] + wave * kC;
    float p = 0.0f;
    for (int i = lane; i < kC; i += 32) p += xs[i] * wrow[i];
#pragma unroll
    for (int off = 16; off > 0; off >>= 1) p += __shfl_down(p, off);
    if (lane == 0) {
      float v = 1.0f / (1.0f + expf(-(p + Bs4[g][wave] + kGateBias)));
      if (g == 1) v *= kMaxLR;
      Os[g][(size_t)t * kH + wave] = v;
    }
  }
}

// ------------------------------------------------------------------
// Init per-(b,h) carry: W1/W2 broadcast from init, S1/S2 = 0.
// ------------------------------------------------------------------
__global__ void init_state_kernel(const float* W1i, const float* W2i, float* W1s,
                                  float* W2s, float* S1, float* S2) {
  int i = blockIdx.x * 256 + threadIdx.x;  // total kB*kH*4096 / 4
  int f = i * 4;
  int src = f % (kH * 4096);
  v4f z; z.x = z.y = z.z = z.w = 0.0f;
  *(v4f*)&W1s[f] = *(const v4f*)&W1i[src];
  *(v4f*)&W2s[f] = *(const v4f*)&W2i[src];
  *(v4f*)&S1[f] = z;
  *(v4f*)&S2[f] = z;
}

// ------------------------------------------------------------------
// Per-chunk gradients (W1,W2 frozen): act, err, chain via 3 WMMA matmuls.
// grid = B*H, block = 256, dynamic LDS = 5*kBUF floats.
// ------------------------------------------------------------------
__global__ __launch_bounds__(256) void chunk_grad_kernel(
    const float* __restrict__ kn, const float* __restrict__ vn,
    const float* __restrict__ W1st, const float* __restrict__ W2st,
    float* __restrict__ actb, float* __restrict__ errb, float* __restrict__ chainb,
    int t0) {
  extern __shared__ float lds[];
  float* B0 = lds;             // K  (c x d) -> later err
  float* B1 = lds + kBUF;      // W1 (d x e)
  float* B2 = lds + 2 * kBUF;  // W2 (e x d) -> later act
  float* B3 = lds + 3 * kBUF;  // h  (c x e)
  float* B4 = lds + 4 * kBUF;  // temp
  const int bid = blockIdx.x;
  const int b = bid / kH, hh = bid % kH;
  const int tid = threadIdx.x;
  const size_t sbase = (size_t)(b * kH + hh) * 4096;

  // async-stage K, W1, W2 tiles into LDS
  tile_load_async64(kn + ((size_t)(b * kT + t0) * kH + hh) * 64, (size_t)kH * 64, B0, tid);
  tile_load_async64(W1st + sbase, 64, B1, tid);
  tile_load_async64(W2st + sbase, 64, B2, tid);
  async_join();
  __syncthreads();

  // h[c,e] = sum_d K[c,d] * W2[e,d]
  mm64(B0, B2, B3, nullptr, 0.0f, true);
  __syncthreads();
#pragma unroll
  for (int i = 0; i < 4; ++i) {
    int fidx = (tid + 256 * i) * 4;
    int r = fidx >> 6, c = fidx & 63;
    int o = r * kST + c;
    v4f hv = *(const v4f*)&B3[o];
    v4f av;
    av.x = gelu_f(hv.x); av.y = gelu_f(hv.y); av.z = gelu_f(hv.z); av.w = gelu_f(hv.w);
    *(v4f*)&B2[o] = av;                  // overwrite W2 with act
    *(v4f*)&actb[sbase + fidx] = av;
  }
  __syncthreads();
  // ypred[c,d] = K[c,d] + sum_e act[c,e] * W1[d,e]
  mm64(B2, B1, B4, B0, 1.0f, true);
  __syncthreads();
#pragma unroll
  for (int i = 0; i < 4; ++i) {
    int fidx = (tid + 256 * i) * 4;
    int r = fidx >> 6, c = fidx & 63;
    int o = r * kST + c;
    v4f yp = *(const v4f*)&B4[o];
    v4f vv = *(const v4f*)&vn[((size_t)(b * kT + t0 + r) * kH + hh) * 64 + c];
    v4f e = yp - vv;
    *(v4f*)&B0[o] = e;                   // overwrite K with err
    *(v4f*)&errb[sbase + fidx] = e;
  }
  __syncthreads();
  // pre_chain[c,e] = sum_d err[c,d] * W1[d,e]
  mm64(B0, B1, B4, nullptr, 0.0f, false);
  __syncthreads();
#pragma unroll
  for (int i = 0; i < 4; ++i) {
    int fidx = (tid + 256 * i) * 4;
    int r = fidx >> 6, c = fidx & 63;
    int o = r * kST + c;
    v4f pc = *(const v4f*)&B4[o];
    v4f hv = *(const v4f*)&B3[o];
    v4f ch;
    ch.x = pc.x * gelu_deriv_f(hv.x);
    ch.y = pc.y * gelu_deriv_f(hv.y);
    ch.z = pc.z * gelu_deriv_f(hv.z);
    ch.w = pc.w * gelu_deriv_f(hv.w);
    *(v4f*)&chainb[sbase + fidx] = ch;
  }
}

// ------------------------------------------------------------------
// theta-scan of momentum with window-4 gamma-weighted rank-1 updates.
// grid = 2*B*H (path 0 -> W1, 1 -> W2), block = 256.
// ------------------------------------------------------------------
__global__ __launch_bounds__(256) void scan_S_kernel(
    const float* __restrict__ errb, const float* __restrict__ actb,
    const float* __restrict__ chainb, const float* __restrict__ kn,
    const float* __restrict__ gam, const float* __restrict__ eta,
    const float* __restrict__ theta, float* __restrict__ SW1, float* __restrict__ SW2,
    float* __restrict__ allS1, float* __restrict__ allS2, int t0) {
  const int bid = blockIdx.x;
  const int p = bid / (kB * kH);
  const int r0 = bid % (kB * kH);
  const int b = r0 / kH, hh = r0 % kH;
  const int tid = threadIdx.x;
  const float* Lsrc = (p == 0) ? errb : chainb;  // [b][h][c][64]
  float* Sstate = ((p == 0) ? SW1 : SW2) + (size_t)(b * kH + hh) * 4096;
  float* outS = (p == 0) ? allS1 : allS2;

  __shared__ __attribute__((aligned(16))) float Lr[4][64];
  __shared__ __attribute__((aligned(16))) float Rr[4][64];
  __shared__ float gr[4], sc[2];
  v4f S4[4];
#pragma unroll
  for (int i = 0; i < 4; ++i) S4[i] = *(const v4f*)&Sstate[(tid + 256 * i) * 4];
  const float scale = 2.0f / (float)kD;

  for (int j = 0; j < kCS; ++j) {
    __syncthreads();
    const int slot = j & 3;
    if (tid < 64) {
      Lr[slot][tid] = Lsrc[((size_t)(b * kH + hh) * 64 + j) * 64 + tid];
    } else if (tid < 128) {
      int d = tid - 64;
      Rr[slot][d] = (p == 0) ? actb[((size_t)(b * kH + hh) * 64 + j) * 64 + d]
                             : kn[((size_t)(b * kT + t0 + j) * kH + hh) * 64 + d];
    } else if (tid == 128) {
      int gi = (b * kT + t0 + j) * kH + hh;
      gr[slot] = gam[gi];
      sc[0] = eta[gi];
      sc[1] = theta[gi];
    }
    __syncthreads();
    const float et = sc[0] * scale, th = sc[1];
    const int j0 = (j - 3 < 0) ? 0 : j - 3;
    const size_t obase = ((size_t)((b * kCS + j) * kH + hh)) * 4096;
#pragma unroll
    for (int i = 0; i < 4; ++i) {
      int fidx = (tid + 256 * i) * 4;
      int rr = fidx >> 6, cc = fidx & 63;
      v4f acc; acc.x = acc.y = acc.z = acc.w = 0.0f;
      for (int jj = j0; jj <= j; ++jj) {
        int s2 = jj & 3;
        float gl = gr[s2] * Lr[s2][rr];
        v4f rv = *(const v4f*)&Rr[s2][cc];
        acc += gl * rv;
      }
      S4[i] = th * S4[i] - et * acc;
      *(v4f*)&outS[obase + fidx] = S4[i];
    }
  }
#pragma unroll
  for (int i = 0; i < 4; ++i) *(v4f*)&Sstate[(tid + 256 * i) * 4] = S4[i];
}

// ------------------------------------------------------------------
// polar_express: Frobenius normalize + 5 Newton-Schulz steps on a
// 64x64 fp32 matrix, in place (15 WMMA 64^3 matmuls per block).
// grid = 2*B*CS*H, block = 256, dynamic LDS = 3*kBUF floats.
// ------------------------------------------------------------------
__global__ __launch_bounds__(256) void polar_kernel(float* __restrict__ allS1,
                                                    float* __restrict__ allS2) {
  extern __shared__ float lds[];
  float* X = lds;
  float* U = lds + kBUF;
  float* V = lds + 2 * kBUF;
  __shared__ float red[256];
  const int per = kB * kCS * kH;
  const int bid = blockIdx.x;
  float* base = ((bid < per) ? allS1 : allS2) + (size_t)(bid % per) * 4096;
  const int tid = threadIdx.x;

  tile_load_async64(base, 64, X, tid);
  async_join();
  __syncthreads();

  float ssq = 0.0f;
#pragma unroll
  for (int i = 0; i < 4; ++i) {
    int fidx = (tid + 256 * i) * 4;
    v4f v = *(const v4f*)&X[(fidx >> 6) * kST + (fidx & 63)];
    ssq += v.x * v.x + v.y * v.y + v.z * v.z + v.w * v.w;
  }
  red[tid] = ssq;
  __syncthreads();
#pragma unroll
  for (int s = 128; s > 0; s >>= 1) {
    if (tid < s) red[tid] += red[tid + s];
    __syncthreads();
  }
  const float frob = sqrtf(red[0] + 1e-12f);
  const float inv = 1.0f / (frob * 1.01f + 1e-6f);
  __syncthreads();
#pragma unroll
  for (int i = 0; i < 4; ++i) {
    int fidx = (tid + 256 * i) * 4;
    int o = (fidx >> 6) * kST + (fidx & 63);
    v4f v = *(const v4f*)&X[o];
    *(v4f*)&X[o] = v * inv;
  }

  for (int s = 0; s < 5; ++s) {
    const float a = PE_A[s], pb = PE_B[s], pc = PE_C[s];
    __syncthreads();
    mm64(X, X, U, nullptr, 0.0f, true);    // U = X * X^T
    __syncthreads();
    mm64(U, U, V, nullptr, 0.0f, false);   // V = U * U
    __syncthreads();
#pragma unroll
    for (int i = 0; i < 4; ++i) {
      int fidx = (tid + 256 * i) * 4;
      int o = (fidx >> 6) * kST + (fidx & 63);
      v4f uv = *(const v4f*)&U[o];
      v4f vv = *(const v4f*)&V[o];
      *(v4f*)&U[o] = pb * uv + pc * vv;    // Bm
    }
    __syncthreads();
    mm64(U, X, V, X, a, false);            // V = a*X + Bm*X
    float* t = X; X = V; V = t;            // rotate
  }
  __syncthreads();
#pragma unroll
  for (int i = 0; i < 4; ++i) {
    int fidx = (tid + 256 * i) * 4;
    *(v4f*)&base[fidx] = *(const v4f*)&X[(fidx >> 6) * kST + (fidx & 63)];
  }
}

// ------------------------------------------------------------------
// alpha-scan over polar outputs (in place) + carry W state out.
// grid = 2*B*H, block = 256.
// ------------------------------------------------------------------
__global__ __launch_bounds__(256) void alpha_scan_kernel(float* __restrict__ allS1,
                                                         float* __restrict__ allS2,
                                                         float* __restrict__ W1st,
                                                         float* __restrict__ W2st,
                                                         const float* __restrict__ alpha,
                                                         int t0) {
  const int bid = blockIdx.x;
  const int p = bid / (kB * kH);
  const int r0 = bid % (kB * kH);
  const int b = r0 / kH, hh = r0 % kH;
  const int tid = threadIdx.x;
  float* Wst = ((p == 0) ? W1st : W2st) + (size_t)(b * kH + hh) * 4096;
  float* Xb = (p == 0) ? allS1 : allS2;

  v4f h4[4];
#pragma unroll
  for (int i = 0; i < 4; ++i) h4[i] = *(const v4f*)&Wst[(tid + 256 * i) * 4];
  for (int j = 0; j < kCS; ++j) {
    const float al = alpha[(size_t)(b * kT + t0 + j) * kH + hh];
    const size_t base = ((size_t)((b * kCS + j) * kH + hh)) * 4096;
#pragma unroll
    for (int i = 0; i < 4; ++i) {
      int fidx = (tid + 256 * i) * 4;
      v4f x = *(const v4f*)&Xb[base + fidx];
      h4[i] = al * h4[i] + x;
      *(v4f*)&Xb[base + fidx] = h4[i];
    }
  }
#pragma unroll
  for (int i = 0; i < 4; ++i) *(v4f*)&Wst[(tid + 256 * i) * 4] = h4[i];
}

// ------------------------------------------------------------------
// y = q + W1[t] @ gelu(W2[t] @ q) per timestep. grid = B*CS*H, block = 64.
// ------------------------------------------------------------------
__global__ void out_kernel(const float* __restrict__ qn, const float* __restrict__ allW1,
                           const float* __restrict__ allW2, float* __restrict__ yall,
                           int t0) {
  const int bid = blockIdx.x;
  const int hh = bid % kH;
  const int c = (bid / kH) % kCS;
  const int b = bid / (kH * kCS);
  const int d = threadIdx.x;
  __shared__ __attribute__((aligned(16))) float qs[64];
  __shared__ __attribute__((aligned(16))) float ha[64];
  const size_t qoff = ((size_t)(b * kT + t0 + c) * kH + hh) * 64;
  qs[d] = qn[qoff + d];
  __syncthreads();
  const size_t mbase = ((size_t)((b * kCS + c) * kH + hh)) * 4096;
  float acc = 0.0f;
#pragma unroll 4
  for (int dd = 0; dd < 64; dd += 4) {
    v4f w = *(const v4f*)&allW2[mbase + d * 64 + dd];
    v4f qv = *(const v4f*)&qs[dd];
    acc += w.x * qv.x + w.y * qv.y + w.z * qv.z + w.w * qv.w;
  }
  ha[d] = gelu_f(acc);
  __syncthreads();
  float acc2 = 0.0f;
#pragma unroll 4
  for (int ee = 0; ee < 64; ee += 4) {
    v4f w = *(const v4f*)&allW1[mbase + d * 64 + ee];
    v4f hv = *(const v4f*)&ha[ee];
    acc2 += w.x * hv.x + w.y * hv.y + w.z * hv.z + w.w * hv.w;
  }
  yall[qoff + d] = qs[d] + acc2;
}

// ==================================================================
extern "C" void kernel_launch(void* const* d_in, const int* in_sizes, int n_in,
                              void* d_out, int out_size, void* d_ws, size_t ws_size,
                              hipStream_t stream) {
  const float* x       = (const float*)d_in[0];
  const float* c_q_w   = (const float*)d_in[1];
  const float* c_k_w   = (const float*)d_in[2];
  const float* c_v_w   = (const float*)d_in[3];
  const float* c_proj_w= (const float*)d_in[4];
  const float* conv_q_w= (const float*)d_in[5];
  const float* conv_q_b= (const float*)d_in[6];
  const float* conv_k_w= (const float*)d_in[7];
  const float* conv_k_b= (const float*)d_in[8];
  const float* conv_v_w= (const float*)d_in[9];
  const float* conv_v_b= (const float*)d_in[10];
  const float* ga_w = (const float*)d_in[11];
  const float* ga_b = (const float*)d_in[12];
  const float* ge_w = (const float*)d_in[13];
  const float* ge_b = (const float*)d_in[14];
  const float* gt_w = (const float*)d_in[15];
  const float* gt_b = (const float*)d_in[16];
  const float* gg_w = (const float*)d_in[17];
  const float* gg_b = (const float*)d_in[18];
  const float* poly = (const float*)d_in[19];
  const float* W1i  = (const float*)d_in[20];
  const float* W2i  = (const float*)d_in[21];
  float* out = (float*)d_out;

  // ---- workspace carve-up (floats) ----
  float* ws = (float*)d_ws;
  const size_t BT = (size_t)kB * kT;                 // 1024
  size_t off = 0;
  float* qraw = ws + off; off += BT * kHD;
  float* kraw = ws + off; off += BT * kHD;
  float* vraw = ws + off; off += BT * kHD;
  float* qn   = ws + off; off += BT * kHD;
  float* kn   = ws + off; off += BT * kHD;
  float* vn   = ws + off; off += BT * kHD;
  float* alpha= ws + off; off += BT * kH;
  float* eta  = ws + off; off += BT * kH;
  float* theta= ws + off; off += BT * kH;
  float* gamma= ws + off; off += BT * kH;
  float* W1st = ws + off; off += (size_t)kB * kH * 4096;
  float* W2st = ws + off; off += (size_t)kB * kH * 4096;
  float* SW1  = ws + off; off += (size_t)kB * kH * 4096;
  float* SW2  = ws + off; off += (size_t)kB * kH * 4096;
  float* actb = ws + off; off += (size_t)kB * kH * kCS * 64;
  float* errb = ws + off; off += (size_t)kB * kH * kCS * 64;
  float* chainb = ws + off; off += (size_t)kB * kH * kCS * 64;
  float* allS1 = ws + off; off += (size_t)kB * kCS * kH * 4096;
  float* allS2 = ws + off; off += (size_t)kB * kCS * kH * 4096;
  float* yall  = ws + off; off += BT * kHD;
  (void)ws_size; (void)in_sizes; (void)n_in; (void)out_size;

  const size_t lds3 = 3 * kBUF * sizeof(float);
  const size_t lds4 = 4 * kBUF * sizeof(float);
  const size_t lds5 = 5 * kBUF * sizeof(float);

  // 1) carry init
  init_state_kernel<<<(kB * kH * 4096 / 4) / 256, 256, 0, stream>>>(W1i, W2i, W1st, W2st,
                                                                    SW1, SW2);
  // 2) QKV projections (f32 WMMA GEMM, async double-buffered)
  {
    dim3 g(BT / 64, kHD / 64);
    gemm_xwt_kernel<<<g, 256, lds4, stream>>>(x, c_q_w, qraw, (int)BT, kHD, kC);
    gemm_xwt_kernel<<<g, 256, lds4, stream>>>(x, c_k_w, kraw, (int)BT, kHD, kC);
    gemm_xwt_kernel<<<g, 256, lds4, stream>>>(x, c_v_w, vraw, (int)BT, kHD, kC);
  }
  // 3) gates
  gates_kernel<<<(int)BT, 256, 0, stream>>>(x, ga_w, ga_b, ge_w, ge_b, gt_w, gt_b, gg_w,
                                            gg_b, alpha, eta, theta, gamma);
  // 4) conv + poly + rmsnorm
  convhead_kernel<<<kB * kT * kH, 64, 0, stream>>>(qraw, conv_q_w, conv_q_b, poly, qn, 1);
  convhead_kernel<<<kB * kT * kH, 64, 0, stream>>>(kraw, conv_k_w, conv_k_b, poly, kn, 1);
  convhead_kernel<<<kB * kT * kH, 64, 0, stream>>>(vraw, conv_v_w, conv_v_b, poly, vn, 0);

  // 5) chunked recurrence
  for (int chunk = 0; chunk < kNC; ++chunk) {
    const int t0 = chunk * kCS;
    chunk_grad_kernel<<<kB * kH, 256, lds5, stream>>>(kn, vn, W1st, W2st, actb, errb,
                                                      chainb, t0);
    scan_S_kernel<<<2 * kB * kH, 256, 0, stream>>>(errb, actb, chainb, kn, gamma, eta,
                                                   theta, SW1, SW2, allS1, allS2, t0);
    polar_kernel<<<2 * kB * kCS * kH, 256, lds3, stream>>>(allS1, allS2);
    alpha_scan_kernel<<<2 * kB * kH, 256, 0, stream>>>(allS1, allS2, W1st, W2st, alpha,
                                                       t0);
    out_kernel<<<kB * kCS * kH, 64, 0, stream>>>(qn, allS1, allS2, yall, t0);
  }

  // 6) output projection
  {
    dim3 g(BT / 64, kC / 64);
    gemm_xwt_kernel<<<g, 256, lds4, stream>>>(yall, c_proj_w, out, (int)BT, kC, kHD);
  }
}